// MultiboxLoss_19155554140648
// MI455X (gfx1250) — compile-verified
//
#include <hip/hip_runtime.h>
#include <math.h>

#define NBATCH 64
#define NOBJ 16
#define NPRI 8732
#define NCLS 21
#define THRESH 0.5f

#ifndef __has_builtin
#define __has_builtin(x) 0
#endif

#if __has_builtin(__builtin_amdgcn_global_load_async_to_lds_b128)
#define HAVE_ASYNC_LDS 1
#else
#define HAVE_ASYNC_LDS 0
#endif

typedef int v4i __attribute__((ext_vector_type(4)));
typedef __attribute__((address_space(1))) v4i g_v4i;  // global (prints as __device__)
typedef __attribute__((address_space(3))) v4i l_v4i;  // LDS (prints as __shared__)

// 16B global -> LDS async copy (CDNA5 GLOBAL_LOAD_ASYNC_TO_LDS_B128, ASYNCcnt)
__device__ __forceinline__ void async_g2l_b128(const float* g, float* l) {
#if HAVE_ASYNC_LDS
    __builtin_amdgcn_global_load_async_to_lds_b128((g_v4i*)g, (l_v4i*)l,
                                                   /*offset=*/0, /*cpol=*/0);
#else
    *(float4*)l = *(const float4*)g;
#endif
}

__device__ __forceinline__ void wait_async_lds() {
#if HAVE_ASYNC_LDS
#if __has_builtin(__builtin_amdgcn_s_wait_asynccnt)
    __builtin_amdgcn_s_wait_asynccnt(0);
#else
    asm volatile("s_wait_asynccnt 0x0" ::: "memory");
#endif
#endif
}

// ---------------------------------------------------------------------------
// Kernel 0: zero the global accumulators (ws is poisoned by harness).
// ---------------------------------------------------------------------------
__global__ void init_kernel(float* acc) {
    if (threadIdx.x < 4) acc[threadIdx.x] = 0.0f;
}

// ---------------------------------------------------------------------------
// Kernel 1: per-batch prior<->object matching.
// One workgroup (1024 threads = 32 wave32) per batch.
// ---------------------------------------------------------------------------
__global__ __launch_bounds__(1024)
void match_kernel(const float* __restrict__ boxes,     // [B, M, 4] xyxy
                  const int*   __restrict__ labels,    // [B, M]
                  const float* __restrict__ prior,     // [P, 4] cxcywh
                  int*   __restrict__ objm_g,          // [B, P]
                  int*   __restrict__ labp_g,          // [B, P]
                  int*   __restrict__ npos_g)          // [B]
{
    __shared__ float s_bx[NOBJ][4];
    __shared__ float s_barea[NOBJ];
    __shared__ float s_pmax[NPRI];            // per-prior best IoU
    __shared__ unsigned char s_pobj[NPRI];    // per-prior best object
    __shared__ float s_wval[NOBJ][32];
    __shared__ int   s_widx[NOBJ][32];
    __shared__ int   s_pfo[NOBJ];             // prior_for_obj
    __shared__ int   s_cnt[32];

    const int b = blockIdx.x;
    const int tid = threadIdx.x;
    const int lane = tid & 31;
    const int wv = tid >> 5;

    if (tid < NOBJ) {
        const float* bb = &boxes[((size_t)b * NOBJ + tid) * 4];
        float x0 = bb[0], y0 = bb[1], x1 = bb[2], y1 = bb[3];
        s_bx[tid][0] = x0; s_bx[tid][1] = y0;
        s_bx[tid][2] = x1; s_bx[tid][3] = y1;
        s_barea[tid] = (x1 - x0) * (y1 - y0);
    }
    __syncthreads();

    // per-thread running best prior for each object (lowest p on ties via strict >)
    float bmv[NOBJ];
    int   bmi[NOBJ];
#pragma unroll
    for (int m = 0; m < NOBJ; ++m) { bmv[m] = -1.0f; bmi[m] = 0; }

    for (int p = tid; p < NPRI; p += 1024) {
        const float* pc = &prior[(size_t)p * 4];
        float cx = pc[0], cy = pc[1], pw = pc[2], ph = pc[3];
        float px0 = cx - pw * 0.5f, py0 = cy - ph * 0.5f;
        float px1 = cx + pw * 0.5f, py1 = cy + ph * 0.5f;
        float parea = (px1 - px0) * (py1 - py0);

        float best = -1.0f;
        int bestm = 0;
#pragma unroll
        for (int m = 0; m < NOBJ; ++m) {
            float tlx = fmaxf(s_bx[m][0], px0);
            float tly = fmaxf(s_bx[m][1], py0);
            float brx = fminf(s_bx[m][2], px1);
            float bry = fminf(s_bx[m][3], py1);
            float w = fmaxf(brx - tlx, 0.0f);
            float h = fmaxf(bry - tly, 0.0f);
            float inter = w * h;
            float iouv = inter / (s_barea[m] + parea - inter);
            if (iouv > best) { best = iouv; bestm = m; }      // first-m semantics
            if (iouv > bmv[m]) { bmv[m] = iouv; bmi[m] = p; } // first-p semantics
        }
        s_pmax[p] = best;
        s_pobj[p] = (unsigned char)bestm;
    }

    // reduce per-object argmax across the workgroup (ties -> lowest prior index)
#pragma unroll
    for (int m = 0; m < NOBJ; ++m) {
        float v = bmv[m];
        int   ix = bmi[m];
        for (int off = 16; off > 0; off >>= 1) {
            float ov = __shfl_down(v, off, 32);
            int   oi = __shfl_down(ix, off, 32);
            if (ov > v || (ov == v && oi < ix)) { v = ov; ix = oi; }
        }
        if (lane == 0) { s_wval[m][wv] = v; s_widx[m][wv] = ix; }
    }
    __syncthreads();
    if (tid < NOBJ) {
        float v = s_wval[tid][0];
        int ix = s_widx[tid][0];
        for (int w = 1; w < 32; ++w) {
            float ov = s_wval[tid][w];
            int   oi = s_widx[tid][w];
            if (ov > v || (ov == v && oi < ix)) { v = ov; ix = oi; }
        }
        s_pfo[tid] = ix;
    }
    __syncthreads();
    // bidirectional fix: sequential, last-wins on duplicate priors
    if (tid == 0) {
        for (int m = 0; m < NOBJ; ++m) {
            int pf = s_pfo[m];
            s_pobj[pf] = (unsigned char)m;
            s_pmax[pf] = 1.0f;
        }
    }
    __syncthreads();

    int cnt = 0;
    for (int p = tid; p < NPRI; p += 1024) {
        int om = (int)s_pobj[p];
        int lbl = (s_pmax[p] < THRESH) ? 0 : labels[b * NOBJ + om];
        size_t gi = (size_t)b * NPRI + p;
        objm_g[gi] = om;
        labp_g[gi] = lbl;
        cnt += (lbl != 0) ? 1 : 0;
    }
    for (int off = 16; off > 0; off >>= 1) cnt += __shfl_down(cnt, off, 32);
    if (lane == 0) s_cnt[wv] = cnt;
    __syncthreads();
    if (tid == 0) {
        int t = 0;
        for (int w = 0; w < 32; ++w) t += s_cnt[w];
        npos_g[b] = t;
    }
}

// ---------------------------------------------------------------------------
// Kernel 2: per-prior cross-entropy + positive-CE sum + SmoothL1 loc loss.
// 256 threads, 256 priors per block, scores staged to LDS via async B128.
// ---------------------------------------------------------------------------
__global__ __launch_bounds__(256)
void ce_kernel(const float* __restrict__ pre_score,  // [B, P, C]
               const float* __restrict__ pre_box,    // [B, P, 4]
               const float* __restrict__ boxes,      // [B, M, 4]
               const float* __restrict__ prior,      // [P, 4]
               const int*   __restrict__ objm_g,
               const int*   __restrict__ labp_g,
               float* __restrict__ ceneg,            // [B, P]
               float* __restrict__ acc)              // [0]=posi_ce [1]=loc
{
    __shared__ __align__(16) float s_sc[256 * NCLS];  // 21504 B, stride 21 conflict-free
    __shared__ float s_red[2][8];

    const int b = blockIdx.y;
    const int p0 = blockIdx.x * 256;
    const int np = min(256, NPRI - p0);
    const size_t base = ((size_t)b * NPRI + p0) * NCLS;

    // CDNA5: prefetch the pre_box tile we may touch later (global_prefetch_b8)
    __builtin_prefetch(&pre_box[((size_t)b * NPRI + p0) * 4], 0, 3);

    // async global->LDS staging: 16B chunks (base and s_sc are 16B aligned;
    // tot = 21*np is divisible by 4 for both np=256 and the tail np=28)
    const int tot = np * NCLS;
    const int nch = tot >> 2;
    for (int c = threadIdx.x; c < nch; c += 256)
        async_g2l_b128(&pre_score[base + (size_t)c * 4], &s_sc[c * 4]);
    for (int i = (nch << 2) + threadIdx.x; i < tot; i += 256)
        s_sc[i] = pre_score[base + i];   // remainder guard (not taken for this shape)
    wait_async_lds();
    __syncthreads();

    float posi_ce = 0.0f, loc = 0.0f;
    const int t = threadIdx.x;
    if (t < np) {
        const int p = p0 + t;
        const float* s = &s_sc[t * NCLS];
        float mx = s[0];
#pragma unroll
        for (int c = 1; c < NCLS; ++c) mx = fmaxf(mx, s[c]);
        float sum = 0.0f;
#pragma unroll
        for (int c = 0; c < NCLS; ++c) sum += __expf(s[c] - mx);
        float lse = mx + __logf(sum);

        const size_t gi = (size_t)b * NPRI + p;
        const int lbl = labp_g[gi];
        const float ce = lse - s[lbl];
        const bool pos = (lbl != 0);
        ceneg[gi] = pos ? 0.0f : ce;

        if (pos) {
            posi_ce = ce;
            const int om = objm_g[gi];
            const float* bx = &boxes[((size_t)b * NOBJ + om) * 4];
            float x0 = bx[0], y0 = bx[1], x1 = bx[2], y1 = bx[3];
            float cx = (x0 + x1) * 0.5f, cy = (y0 + y1) * 0.5f;
            float w = x1 - x0, h = y1 - y0;
            const float* pr = &prior[(size_t)p * 4];
            float g0 = (cx - pr[0]) / (pr[2] / 10.0f);
            float g1 = (cy - pr[1]) / (pr[3] / 10.0f);
            float g2 = __logf(w / pr[2]) * 5.0f;
            float g3 = __logf(h / pr[3]) * 5.0f;
            float g[4] = { g0, g1, g2, g3 };
            const float* pb = &pre_box[gi * 4];
#pragma unroll
            for (int k = 0; k < 4; ++k) {
                float d = pb[k] - g[k];
                float ad = fabsf(d);
                loc += (ad < 1.0f) ? 0.5f * d * d : ad - 0.5f;
            }
        }
    }

    const int lane = threadIdx.x & 31, wv = threadIdx.x >> 5;
    for (int off = 16; off > 0; off >>= 1) {
        posi_ce += __shfl_down(posi_ce, off, 32);
        loc     += __shfl_down(loc, off, 32);
    }
    if (lane == 0) { s_red[0][wv] = posi_ce; s_red[1][wv] = loc; }
    __syncthreads();
    if (threadIdx.x == 0) {
        float a = 0.0f, l = 0.0f;
        for (int w = 0; w < 8; ++w) { a += s_red[0][w]; l += s_red[1][w]; }
        if (a != 0.0f) atomicAdd(acc + 0, a);
        if (l != 0.0f) atomicAdd(acc + 1, l);
    }
}

// ---------------------------------------------------------------------------
// Kernel 3: per-batch hard-negative mining: sum of top-K (K = 3*n_pos) of
// conf_neg via radix-select on float bit patterns (CE >= 0 -> uint order).
// One workgroup per batch, values LDS-resident (async-staged).
// ---------------------------------------------------------------------------
__global__ __launch_bounds__(1024)
void topk_kernel(const float* __restrict__ ceneg,
                 const int*   __restrict__ npos_g,
                 float* __restrict__ acc)
{
    __shared__ __align__(16) float s_v[NPRI];
    __shared__ int   s_cnt;
    __shared__ float s_fsum;

    const int b = blockIdx.x;
    const int tid = threadIdx.x;
    const int lane = tid & 31;

    // async global->LDS staging (NPRI = 8732 divisible by 4; base 16B aligned)
    for (int c = tid; c < (NPRI >> 2); c += 1024)
        async_g2l_b128(&ceneg[(size_t)b * NPRI + (size_t)c * 4], &s_v[c * 4]);

    int K = 3 * npos_g[b];
    if (K > NPRI) K = NPRI;
    wait_async_lds();
    __syncthreads();
    if (K <= 0) return;  // uniform across block (npos is per-batch)

    // find largest T with count(bits >= T) >= K  ->  T = bits of K-th largest
    unsigned lo = 0u, hi = 0x7F800001u;
    while (hi - lo > 1u) {
        unsigned mid = lo + ((hi - lo) >> 1u);
        if (tid == 0) s_cnt = 0;
        __syncthreads();
        int c = 0;
        for (int p = tid; p < NPRI; p += 1024)
            c += (__float_as_uint(s_v[p]) >= mid) ? 1 : 0;
        for (int off = 16; off > 0; off >>= 1) c += __shfl_down(c, off, 32);
        if (lane == 0) atomicAdd(&s_cnt, c);
        __syncthreads();
        int cnt = s_cnt;
        __syncthreads();  // protect s_cnt before next-iteration zeroing
        if (cnt >= K) lo = mid; else hi = mid;
    }
    const float tval = __uint_as_float(lo);

    if (tid == 0) { s_cnt = 0; s_fsum = 0.0f; }
    __syncthreads();
    float ls = 0.0f;
    int lc = 0;
    for (int p = tid; p < NPRI; p += 1024) {
        float v = s_v[p];
        if (__float_as_uint(v) > lo) { ls += v; ++lc; }
    }
    for (int off = 16; off > 0; off >>= 1) {
        ls += __shfl_down(ls, off, 32);
        lc += __shfl_down(lc, off, 32);
    }
    if (lane == 0) { atomicAdd(&s_fsum, ls); atomicAdd(&s_cnt, lc); }
    __syncthreads();
    if (tid == 0) {
        float total = s_fsum + (float)(K - s_cnt) * tval;  // ties at threshold
        atomicAdd(acc + 2, total);
    }
}

// ---------------------------------------------------------------------------
// Kernel 4: combine into the scalar loss.
// ---------------------------------------------------------------------------
__global__ void final_kernel(const int* __restrict__ npos_g,
                             const float* __restrict__ acc,
                             float* __restrict__ out)
{
    if (threadIdx.x == 0) {
        int npt = 0;
        for (int b = 0; b < NBATCH; ++b) npt += npos_g[b];
        float fnpt = (float)npt;
        float conf = (acc[0] + acc[2]) / fnpt;
        float locl = acc[1] / (fnpt * 4.0f);
        out[0] = conf + locl;
    }
}

// ---------------------------------------------------------------------------
extern "C" void kernel_launch(void* const* d_in, const int* in_sizes, int n_in,
                              void* d_out, int out_size, void* d_ws, size_t ws_size,
                              hipStream_t stream) {
    (void)in_sizes; (void)n_in; (void)out_size; (void)ws_size;

    const float* pre_box   = (const float*)d_in[0];
    const float* pre_score = (const float*)d_in[1];
    const float* boxes     = (const float*)d_in[2];
    const int*   labels    = (const int*)d_in[3];
    const float* prior     = (const float*)d_in[4];
    float* out = (float*)d_out;

    // workspace layout
    const size_t BP = (size_t)NBATCH * NPRI;
    float* ceneg = (float*)d_ws;
    int*   objm  = (int*)(ceneg + BP);
    int*   labp  = (int*)(objm + BP);
    int*   npos  = (int*)(labp + BP);
    float* acc   = (float*)(npos + NBATCH);  // [0]=posi_ce [1]=loc [2]=hard_neg

    init_kernel<<<1, 64, 0, stream>>>(acc);

    match_kernel<<<NBATCH, 1024, 0, stream>>>(boxes, labels, prior, objm, labp, npos);

    dim3 g2((NPRI + 255) / 256, NBATCH);
    ce_kernel<<<g2, 256, 0, stream>>>(pre_score, pre_box, boxes, prior,
                                      objm, labp, ceneg, acc);

    topk_kernel<<<NBATCH, 1024, 0, stream>>>(ceneg, npos, acc);

    final_kernel<<<1, 64, 0, stream>>>(npos, acc, out);
}